// GatedFeatureRead_84997402788506
// MI455X (gfx1250) — compile-verified
//
#include <hip/hip_runtime.h>
#include <math.h>

typedef __attribute__((ext_vector_type(2))) float v2f;
typedef __attribute__((ext_vector_type(8))) float v8f;

#define B_SZ 1024
#define C_SZ 256
#define DF   512
#define DT   512
#define HID  32

// ---------------------------------------------------------------------------
// GEMM1: Qt[b,f] = sum_t Q[b,t] * Wk[t,f]      (M=1024, N=512, K=512, fp32)
// One wave computes one 16x16 output tile with V_WMMA_F32_16X16X4_F32.
// A (16x4, MxK):  lane = m + 16*(k>=2), vgpr = k&1   (ISA 7.12.2)
// B (4x16, KxN):  lane = n + 16*(k>=2), vgpr = k&1
// C/D (16x16):    vgpr r -> M = r + 8*(lane>=16), N = lane&15
// ---------------------------------------------------------------------------
__global__ void __launch_bounds__(128)
k_qtilde(const float* __restrict__ Q, const float* __restrict__ Wk,
         float* __restrict__ Qt)
{
    const int lane  = threadIdx.x & 31;
    const int wave  = (blockIdx.x << 2) | (threadIdx.x >> 5);
    const int tilesN = DF / 16;                 // 32
    const int tm = wave / tilesN;               // 0..63
    const int tn = wave % tilesN;               // 0..31
    const int mrow  = tm * 16 + (lane & 15);
    const int ncol  = tn * 16 + (lane & 15);
    const int khalf = (lane >> 4) << 1;         // 0 or 2

    const float* qrow = Q + (size_t)mrow * DT;
    v8f acc = {};
    for (int k0 = 0; k0 < DT; k0 += 4) {
        v2f a = *(const v2f*)(qrow + k0 + khalf);              // K = k0+khalf, +1
        v2f b;
        b.x = Wk[(size_t)(k0 + khalf)     * DF + ncol];
        b.y = Wk[(size_t)(k0 + khalf + 1) * DF + ncol];
        acc = __builtin_amdgcn_wmma_f32_16x16x4_f32(
                  false, a, false, b, (short)0, acc, false, false);
    }
    const int mbase = tm * 16 + ((lane >> 4) << 3);
#pragma unroll
    for (int r = 0; r < 8; ++r)
        Qt[(size_t)(mbase + r) * DF + ncol] = acc[r];
}

// ---------------------------------------------------------------------------
// GEMM2: out[b,t] = gate[b] * ( sum_f FR[b,f] * Wv[t,f] + Wv_b[t] )
// Same tiling; B-matrix element [k,n] = Wv[n*DF + k] -> contiguous float2.
// ---------------------------------------------------------------------------
__global__ void __launch_bounds__(128)
k_out(const float* __restrict__ FR, const float* __restrict__ Wv,
      const float* __restrict__ Wv_b, const float* __restrict__ gate,
      float* __restrict__ out)
{
    const int lane  = threadIdx.x & 31;
    const int wave  = (blockIdx.x << 2) | (threadIdx.x >> 5);
    const int tilesN = DT / 16;                 // 32
    const int tm = wave / tilesN;
    const int tn = wave % tilesN;
    const int mrow  = tm * 16 + (lane & 15);
    const int ncol  = tn * 16 + (lane & 15);
    const int khalf = (lane >> 4) << 1;

    const float* arow = FR + (size_t)mrow * DF;
    const float* brow = Wv + (size_t)ncol * DF;   // row t = ncol, contiguous in f
    v8f acc = {};
    for (int k0 = 0; k0 < DF; k0 += 4) {
        v2f a = *(const v2f*)(arow + k0 + khalf);
        v2f b = *(const v2f*)(brow + k0 + khalf);
        acc = __builtin_amdgcn_wmma_f32_16x16x4_f32(
                  false, a, false, b, (short)0, acc, false, false);
    }
    const int   mbase = tm * 16 + ((lane >> 4) << 3);
    const float bias  = Wv_b[ncol];
#pragma unroll
    for (int r = 0; r < 8; ++r) {
        const int m = mbase + r;
        out[(size_t)m * DT + ncol] = gate[m] * (acc[r] + bias);
    }
}

// ---------------------------------------------------------------------------
// Attention + gate kernel: one workgroup (256 threads) per batch element b.
//  pass1: scores[c] = (q~ . f[b,c]) / sqrt(DT); accumulate sum|f| for scale
//  softmax over C=256 in LDS
//  pass2: FR[b,f] = sum_c w[c] * f[b,c,f]   (coalesced; slab resident in L2)
//  wave 0: exact-gelu gate MLP -> gate[b] (workspace + output tail)
// ---------------------------------------------------------------------------
__global__ void __launch_bounds__(256)
k_attn(const float* __restrict__ Q, const float* __restrict__ features,
       const float* __restrict__ Qt,
       const float* __restrict__ g1_w, const float* __restrict__ g1_b,
       const float* __restrict__ g2_w, const float* __restrict__ g2_b,
       float* __restrict__ FR, float* __restrict__ gate_ws,
       float* __restrict__ gate_out)
{
    __shared__ float s_q[DF];
    __shared__ float s_w[C_SZ];
    __shared__ float s_red[C_SZ];
    __shared__ float s_scale;

    const int b   = blockIdx.x;
    const int tid = threadIdx.x;
    const float* feat = features + (size_t)b * C_SZ * DF;

    for (int i = tid; i < DF; i += 256)
        s_q[i] = Qt[(size_t)b * DF + i];
    __syncthreads();

    // ---- pass 1: one channel per thread -----------------------------------
    const float* frow = feat + (size_t)tid * DF;
    float dot = 0.f, asum = 0.f;
    for (int f = 0; f < DF; f += 4) {
        float4 v = *(const float4*)(frow + f);
        dot  = fmaf(s_q[f+0], v.x, dot);
        dot  = fmaf(s_q[f+1], v.y, dot);
        dot  = fmaf(s_q[f+2], v.z, dot);
        dot  = fmaf(s_q[f+3], v.w, dot);
        asum += fabsf(v.x) + fabsf(v.y) + fabsf(v.z) + fabsf(v.w);
    }
    s_w[tid] = dot * 0.04419417382415922f;      // 1/sqrt(512); Wk_b cancels in softmax

    // mean(|features|) block reduction
    s_red[tid] = asum;
    __syncthreads();
    for (int s = 128; s > 0; s >>= 1) {
        if (tid < s) s_red[tid] += s_red[tid + s];
        __syncthreads();
    }
    if (tid == 0) s_scale = s_red[0] * (1.0f / (float)(C_SZ * DF));
    __syncthreads();

    // ---- softmax over 256 channels ----------------------------------------
    s_red[tid] = s_w[tid];
    __syncthreads();
    for (int s = 128; s > 0; s >>= 1) {
        if (tid < s) s_red[tid] = fmaxf(s_red[tid], s_red[tid + s]);
        __syncthreads();
    }
    const float mx = s_red[0];
    __syncthreads();
    const float e = __expf(s_w[tid] - mx);
    s_red[tid] = e;
    __syncthreads();
    for (int s = 128; s > 0; s >>= 1) {
        if (tid < s) s_red[tid] += s_red[tid + s];
        __syncthreads();
    }
    const float inv = 1.0f / s_red[0];
    __syncthreads();
    s_w[tid] = e * inv;
    __syncthreads();

    // ---- pass 2: weighted feature reduction (coalesced over f) ------------
    float acc0 = 0.f, acc1 = 0.f;
    for (int c = 0; c < C_SZ; ++c) {
        if (c + 8 < C_SZ)
            __builtin_prefetch(feat + (size_t)(c + 8) * DF + tid, 0, 1);
        const float w = s_w[c];
        acc0 = fmaf(w, feat[(size_t)c * DF + tid],       acc0);
        acc1 = fmaf(w, feat[(size_t)c * DF + tid + 256], acc1);
    }
    FR[(size_t)b * DF + tid]       = acc0;
    FR[(size_t)b * DF + tid + 256] = acc1;

    // ---- gate MLP (wave 0): h = gelu(g1_w @ [Q; scale] + g1_b) ------------
    if (tid < HID) {
        const float* gw = g1_w + tid * (DT + 1);
        const float* qb = Q + (size_t)b * DT;
        float h = g1_b[tid] + gw[DT] * s_scale;
        for (int t = 0; t < DT; t += 4) {
            h = fmaf(gw[t+0], qb[t+0], h);
            h = fmaf(gw[t+1], qb[t+1], h);
            h = fmaf(gw[t+2], qb[t+2], h);
            h = fmaf(gw[t+3], qb[t+3], h);
        }
        const float g = 0.5f * h * (1.0f + erff(h * 0.7071067811865476f));
        s_red[tid] = g * g2_w[tid];
    }
    __syncthreads();
    if (tid == 0) {
        float z = g2_b[0];
        for (int j = 0; j < HID; ++j) z += s_red[j];
        const float gate = 1.0f / (1.0f + __expf(-z));
        gate_ws[b]  = gate;
        gate_out[b] = gate;
    }
}

// ---------------------------------------------------------------------------
extern "C" void kernel_launch(void* const* d_in, const int* in_sizes, int n_in,
                              void* d_out, int out_size, void* d_ws, size_t ws_size,
                              hipStream_t stream)
{
    const float* Q    = (const float*)d_in[0];
    const float* feat = (const float*)d_in[1];
    const float* Wk_w = (const float*)d_in[2];
    // d_in[3] = Wk_b: constant per-b score shift, cancels in softmax
    const float* Wv_w = (const float*)d_in[4];
    const float* Wv_b = (const float*)d_in[5];
    const float* g1w  = (const float*)d_in[6];
    const float* g1b  = (const float*)d_in[7];
    const float* g2w  = (const float*)d_in[8];
    const float* g2b  = (const float*)d_in[9];

    float* Qt   = (float*)d_ws;                      // [1024, 512]
    float* FR   = Qt + (size_t)B_SZ * DF;            // [1024, 512]
    float* gate = FR + (size_t)B_SZ * DF;            // [1024]

    float* out_main = (float*)d_out;                 // [1024, 512] gated read
    float* out_gate = out_main + (size_t)B_SZ * DT;  // [1024, 1]   gate

    // 64x32 = 2048 16x16 tiles, 4 waves (128 threads) per block
    k_qtilde<<<512, 128, 0, stream>>>(Q, Wk_w, Qt);
    k_attn  <<<B_SZ, 256, 0, stream>>>(Q, feat, Qt, g1w, g1b, g2w, g2b,
                                       FR, gate, out_gate);
    k_out   <<<512, 128, 0, stream>>>(FR, Wv_w, Wv_b, gate, out_main);
}